// Transformer_50886772523126
// MI455X (gfx1250) — compile-verified
//
#include <hip/hip_runtime.h>
#include <hip/hip_bf16.h>
#include <math.h>

// ---------------------------------------------------------------------------
// Model dims (compile-time constants from the reference)
// ---------------------------------------------------------------------------
#define BB   4
#define SS   1024
#define VV   32000
#define DD   256
#define HH   8
#define LL   4
#define DHH  32
#define MM   (BB * SS)          // 4096 rows

typedef float v2f __attribute__((ext_vector_type(2)));
typedef float v8f __attribute__((ext_vector_type(8)));

// f32 WMMA: D(16x16,f32) = A(16x4,f32) * B(4x16,f32) + C   (ISA 15.10 op 93)
__device__ __forceinline__ v8f wmma_f32_16x16x4(v2f a, v2f b, v8f c) {
  return __builtin_amdgcn_wmma_f32_16x16x4_f32(false, a, false, b, (short)0, c,
                                               false, false);
}

__device__ __forceinline__ float gelu_exact(float x) {
  return 0.5f * x * (1.0f + erff(x * 0.70710678118654752f));
}

// ---------------------------------------------------------------------------
// CDNA5 async copy: GLOBAL_LOAD_ASYNC_TO_LDS_B128 (ASYNCcnt-tracked, §15.18.3
// op 98). VDST = per-lane LDS byte offset; generic LDS pointers are
// {aperture_hi32, lds_offset_lo32}, so truncation yields the offset.
// ---------------------------------------------------------------------------
__device__ __forceinline__ void async_copy_b128(float* lds_dst,
                                                const float* gsrc) {
  unsigned loff = (unsigned)(uintptr_t)lds_dst;
  unsigned long long gaddr = (unsigned long long)(uintptr_t)gsrc;
  asm volatile("global_load_async_to_lds_b128 %0, %1, off"
               :
               : "v"(loff), "v"(gaddr)
               : "memory");
}

__device__ __forceinline__ void wait_async0() {
  asm volatile("s_wait_asynccnt 0x0" ::: "memory");
}

// ---------------------------------------------------------------------------
// Embedding gather: x[row, :] = emb[X[row], :]
// ---------------------------------------------------------------------------
__global__ __launch_bounds__(256) void embed_kernel(
    const int* __restrict__ X, const float* __restrict__ emb,
    float* __restrict__ x) {
  int row = blockIdx.x;
  int t = threadIdx.x;
  int tok = X[row];
  x[(size_t)row * DD + t] = emb[(size_t)tok * DD + t];
}

// ---------------------------------------------------------------------------
// Per-layer diagonal prep:
//   g[f]  = WQ[h,d,d] * WK[h,d,d] * (1/sqrt(DH))   (folds QK scale)
//   vo[f] = WV[h,d,d] * O[f,f]
// ---------------------------------------------------------------------------
__global__ void prep_diag_kernel(const float* __restrict__ WQ,
                                 const float* __restrict__ WK,
                                 const float* __restrict__ WV,
                                 const float* __restrict__ O,
                                 float* __restrict__ g, float* __restrict__ vo) {
  int f = threadIdx.x;            // 0..255
  int h = f >> 5, d = f & 31;
  size_t diag = ((size_t)h * DHH + d) * DHH + d;
  float wq = WQ[diag], wk = WK[diag], wv = WV[diag];
  float od = O[(size_t)f * DD + f];
  g[f] = wq * wk * 0.17677669529663687f;   // 1/sqrt(32)
  vo[f] = wv * od;
}

// ---------------------------------------------------------------------------
// LayerNorm: one block per (b,s) row; emits xn and qp = xn * g
// ---------------------------------------------------------------------------
__global__ __launch_bounds__(256) void ln_kernel(
    const float* __restrict__ x, const float* __restrict__ lng,
    const float* __restrict__ lnb, const float* __restrict__ qscale,
    float* __restrict__ xn, float* __restrict__ qp) {
  __shared__ float red[256];
  int row = blockIdx.x, t = threadIdx.x;
  float v = x[(size_t)row * DD + t];

  red[t] = v;
  __syncthreads();
  for (int o = 128; o > 0; o >>= 1) {
    if (t < o) red[t] += red[t + o];
    __syncthreads();
  }
  float mean = red[0] * (1.0f / DD);
  __syncthreads();

  float dv = v - mean;
  red[t] = dv * dv;
  __syncthreads();
  for (int o = 128; o > 0; o >>= 1) {
    if (t < o) red[t] += red[t + o];
    __syncthreads();
  }
  float var = red[0] * (1.0f / DD);
  float inv = rsqrtf(var + 1e-5f);

  float xv = dv * inv * lng[t] + lnb[t];
  xn[(size_t)row * DD + t] = xv;
  qp[(size_t)row * DD + t] = xv * qscale[t];
}

// ---------------------------------------------------------------------------
// Flash attention with diagonal weights.
//   scores = Qp(16x32) @ KVn^T, online softmax, Y += P @ KVn, scaled by vo.
// Grid: (S/128, H, B). Block: 256 (8 waves); wave w owns queries
// q0 = blockIdx.x*128 + w*16 .. +15. K/V chunks double-buffered via async
// LDS loads so the next chunk streams in under WMMA + softmax work.
// ---------------------------------------------------------------------------
__global__ __launch_bounds__(256) void attn_kernel(
    const float* __restrict__ Qp, const float* __restrict__ KVn,
    const float* __restrict__ vo, float* __restrict__ Y) {
  __shared__ alignas(16) float lKV[2][16 * 36];     // double-buffered chunk
  __shared__ float lP[8][16 * 17];                  // per-wave P staging

  int t = threadIdx.x;
  int lane = t & 31, wid = t >> 5;
  int lane16 = lane & 15, half = lane >> 4;
  int h = blockIdx.y, b = blockIdx.z;
  int q0 = blockIdx.x * 128 + wid * 16;
  int fbase = h * DHH;

  // Q' A-fragments: A(16x32) as 8 chained 16x4 slabs.
  // A layout: m = lane%16, k = 4*kk + 2*(lane/16) + {0,1}
  v2f aq[8];
  {
    const float* qrow = Qp + ((size_t)(b * SS) + q0 + lane16) * DD + fbase;
#pragma unroll
    for (int kk = 0; kk < 8; ++kk) {
      int k0 = kk * 4 + half * 2;
      aq[kk].x = qrow[k0];
      aq[kk].y = qrow[k0 + 1];
    }
  }

  v8f yacc[2];
  float mrow[8], lrow[8];
#pragma unroll
  for (int r = 0; r < 8; ++r) {
    yacc[0][r] = 0.0f; yacc[1][r] = 0.0f;
    mrow[r] = -3.0e38f; lrow[r] = 0.0f;
  }

  // Prologue: async-stage chunk 0 into buffer 0 (threads 0..127, 1xB128 each).
  int sr = t >> 3, sc4 = (t & 7) << 2;
  if (t < 128) {
    async_copy_b128(&lKV[0][sr * 36 + sc4],
                    KVn + ((size_t)(b * SS) + sr) * DD + fbase + sc4);
  }

  int cur = 0;
  for (int kc = 0; kc < SS; kc += 16) {
    wait_async0();        // my wave's outstanding async loads done
    __syncthreads();      // everyone's chunk data visible; prev reads done
    if (kc + 16 < SS && t < 128) {
      async_copy_b128(&lKV[cur ^ 1][sr * 36 + sc4],
                      KVn + ((size_t)(b * SS) + kc + 16 + sr) * DD + fbase +
                          sc4);
    }
    const float* kv = lKV[cur];

    // S_tile(16x16) = Qp @ K^T  : B[k][n] = K[n][k]
    v8f sacc;
#pragma unroll
    for (int r = 0; r < 8; ++r) sacc[r] = 0.0f;
#pragma unroll
    for (int kk = 0; kk < 8; ++kk) {
      int k0 = kk * 4 + half * 2;
      v2f bk;
      bk.x = kv[lane16 * 36 + k0];
      bk.y = kv[lane16 * 36 + k0 + 1];
      sacc = wmma_f32_16x16x4(aq[kk], bk, sacc);
    }

    // Online softmax. C/D layout: row m = r + 8*half, col n = lane16.
#pragma unroll
    for (int r = 0; r < 8; ++r) {
      float s = sacc[r];
      float rm = s;
      rm = fmaxf(rm, __shfl_xor(rm, 1, 16));
      rm = fmaxf(rm, __shfl_xor(rm, 2, 16));
      rm = fmaxf(rm, __shfl_xor(rm, 4, 16));
      rm = fmaxf(rm, __shfl_xor(rm, 8, 16));
      float mn = fmaxf(mrow[r], rm);
      float pe = __expf(s - mn);
      float rs = pe;
      rs += __shfl_xor(rs, 1, 16);
      rs += __shfl_xor(rs, 2, 16);
      rs += __shfl_xor(rs, 4, 16);
      rs += __shfl_xor(rs, 8, 16);
      float alpha = __expf(mrow[r] - mn);
      lrow[r] = lrow[r] * alpha + rs;
      mrow[r] = mn;
      yacc[0][r] *= alpha;
      yacc[1][r] *= alpha;
      lP[wid][(r + 8 * half) * 17 + lane16] = pe;   // reshape D->A via LDS
    }

    // Y(16x32) += P(16x16) @ V(16x32): 4 K-slabs x 2 N-tiles
#pragma unroll
    for (int kk = 0; kk < 4; ++kk) {
      int k0 = kk * 4 + half * 2;
      v2f ap;
      ap.x = lP[wid][lane16 * 17 + k0];
      ap.y = lP[wid][lane16 * 17 + k0 + 1];
#pragma unroll
      for (int nt = 0; nt < 2; ++nt) {
        v2f bv;
        bv.x = kv[k0 * 36 + nt * 16 + lane16];
        bv.y = kv[(k0 + 1) * 36 + nt * 16 + lane16];
        yacc[nt] = wmma_f32_16x16x4(ap, bv, yacc[nt]);
      }
    }
    cur ^= 1;
  }

  // Normalize + apply fused WVdiag*Odiag scale, store.
#pragma unroll
  for (int nt = 0; nt < 2; ++nt) {
#pragma unroll
    for (int r = 0; r < 8; ++r) {
      int q = q0 + r + 8 * half;
      int f = fbase + nt * 16 + lane16;
      Y[((size_t)(b * SS) + q) * DD + f] = yacc[nt][r] / lrow[r] * vo[f];
    }
  }
}

// ---------------------------------------------------------------------------
// WMMA f32 GEMM: C = act(A[M,K] @ W[K,N] + bias) (+ res)
// 128x128 tile per block; 8 waves, each 32(M) x 64(N) = 2x4 accumulators.
// K chunked by 32 through double-buffered LDS filled with async B128 copies.
// act: 0=none, 1=exact GELU. Grid: (N/128, M/128).
// Requires M%128==0, N%128==0, K%32==0.
// ---------------------------------------------------------------------------
__global__ __launch_bounds__(256) void gemm_kernel(
    const float* __restrict__ A, const float* __restrict__ W,
    const float* __restrict__ bias, const float* __restrict__ res,
    float* __restrict__ C, int M, int N, int K, int act) {
  __shared__ alignas(16) float lA[2][128 * 36];   // [m][k], stride 36
  __shared__ alignas(16) float lB[2][32 * 132];   // [k][n], stride 132

  int t = threadIdx.x;
  int lane = t & 31, wid = t >> 5;
  int lane16 = lane & 15, half = lane >> 4;
  int wm = wid & 3, wn = wid >> 2;
  int tM = blockIdx.y * 128, tN = blockIdx.x * 128;

  // Staging coordinates (4x B128 per thread for each of A and B)
  int ar = t >> 3, ac4 = (t & 7) << 2;     // +32 rows per step of 256 threads
  int br = t >> 5, bc4 = (t & 31) << 2;    // +8 rows per step

  v8f acc[2][4];
#pragma unroll
  for (int mi = 0; mi < 2; ++mi)
#pragma unroll
    for (int ni = 0; ni < 4; ++ni)
#pragma unroll
      for (int r = 0; r < 8; ++r) acc[mi][ni][r] = 0.0f;

  // Prologue: stage K-chunk 0 into buffer 0.
#pragma unroll
  for (int i = 0; i < 4; ++i) {
    async_copy_b128(&lA[0][(ar + i * 32) * 36 + ac4],
                    A + (size_t)(tM + ar + i * 32) * K + ac4);
    async_copy_b128(&lB[0][(br + i * 8) * 132 + bc4],
                    W + (size_t)(br + i * 8) * N + tN + bc4);
  }

  int cur = 0;
  for (int kb = 0; kb < K; kb += 32) {
    wait_async0();
    __syncthreads();
    if (kb + 32 < K) {
      int kn = kb + 32;
#pragma unroll
      for (int i = 0; i < 4; ++i) {
        async_copy_b128(&lA[cur ^ 1][(ar + i * 32) * 36 + ac4],
                        A + (size_t)(tM + ar + i * 32) * K + kn + ac4);
        async_copy_b128(&lB[cur ^ 1][(br + i * 8) * 132 + bc4],
                        W + (size_t)(kn + br + i * 8) * N + tN + bc4);
      }
    }
    const float* la = lA[cur];
    const float* lb = lB[cur];

#pragma unroll
    for (int kk = 0; kk < 8; ++kk) {
      int k0 = kk * 4 + half * 2;
      v2f af[2], bf[4];
#pragma unroll
      for (int mi = 0; mi < 2; ++mi) {
        int row = wm * 32 + mi * 16 + lane16;
        af[mi].x = la[row * 36 + k0];
        af[mi].y = la[row * 36 + k0 + 1];
      }
#pragma unroll
      for (int ni = 0; ni < 4; ++ni) {
        int col = wn * 64 + ni * 16 + lane16;
        bf[ni].x = lb[k0 * 132 + col];
        bf[ni].y = lb[(k0 + 1) * 132 + col];
      }
#pragma unroll
      for (int mi = 0; mi < 2; ++mi)
#pragma unroll
        for (int ni = 0; ni < 4; ++ni)
          acc[mi][ni] = wmma_f32_16x16x4(af[mi], bf[ni], acc[mi][ni]);
    }
    cur ^= 1;
  }

  // Epilogue. C/D layout: row = r + 8*half, col = lane16 within each 16x16.
#pragma unroll
  for (int mi = 0; mi < 2; ++mi) {
#pragma unroll
    for (int ni = 0; ni < 4; ++ni) {
      int col = tN + wn * 64 + ni * 16 + lane16;
      float bv = bias ? bias[col] : 0.0f;
#pragma unroll
      for (int r = 0; r < 8; ++r) {
        int row = tM + wm * 32 + mi * 16 + r + 8 * half;
        float v = acc[mi][ni][r] + bv;
        if (act == 1) v = gelu_exact(v);
        if (res) v += res[(size_t)row * N + col];
        C[(size_t)row * N + col] = v;
      }
    }
  }
}

// ---------------------------------------------------------------------------
// Host orchestration
// ---------------------------------------------------------------------------
extern "C" void kernel_launch(void* const* d_in, const int* in_sizes, int n_in,
                              void* d_out, int out_size, void* d_ws,
                              size_t ws_size, hipStream_t stream) {
  (void)in_sizes; (void)n_in; (void)out_size; (void)ws_size;
  const int* X = (const int*)d_in[0];
  const float* emb = (const float*)d_in[1];
  const float* WQ = (const float*)d_in[2];
  const float* WK = (const float*)d_in[3];
  const float* WV = (const float*)d_in[4];
  const float* O = (const float*)d_in[5];
  const float* lng = (const float*)d_in[6];
  const float* lnb = (const float*)d_in[7];
  const float* w1 = (const float*)d_in[8];
  const float* b1 = (const float*)d_in[9];
  const float* w2 = (const float*)d_in[10];
  const float* b2 = (const float*)d_in[11];
  const float* w3 = (const float*)d_in[12];
  const float* b3 = (const float*)d_in[13];
  const float* logitW = (const float*)d_in[14];
  float* out = (float*)d_out;

  const size_t XD = (size_t)MM * DD;   // 4096*256
  float* x = (float*)d_ws;
  float* xn = x + XD;
  float* qp = xn + XD;
  float* ya = qp + XD;
  float* y1 = ya + XD;
  float* y2 = y1 + XD;
  float* gd = y2 + XD;                 // 256
  float* vo = gd + DD;                 // 256

  embed_kernel<<<MM, 256, 0, stream>>>(X, emb, x);

  for (int i = 0; i < LL; ++i) {
    const size_t hdd = (size_t)HH * DHH * DHH;   // per-layer WQ/WK/WV stride
    const size_t dd = (size_t)DD * DD;           // per-layer O/w1/w2/w3 stride
    prep_diag_kernel<<<1, 256, 0, stream>>>(WQ + i * hdd, WK + i * hdd,
                                            WV + i * hdd, O + i * dd, gd, vo);
    ln_kernel<<<MM, 256, 0, stream>>>(x, lng + i * DD, lnb + i * DD, gd, xn,
                                      qp);
    attn_kernel<<<dim3(SS / 128, HH, BB), 256, 0, stream>>>(qp, xn, vo, ya);
    gemm_kernel<<<dim3(DD / 128, MM / 128), 256, 0, stream>>>(
        ya, w1 + i * dd, b1 + i * DD, nullptr, y1, MM, DD, DD, 1);
    gemm_kernel<<<dim3(DD / 128, MM / 128), 256, 0, stream>>>(
        y1, w2 + i * dd, b2 + i * DD, nullptr, y2, MM, DD, DD, 1);
    gemm_kernel<<<dim3(DD / 128, MM / 128), 256, 0, stream>>>(
        y2, w3 + i * dd, b3 + i * DD, x, x, MM, DD, DD, 0);
  }

  // Final logits: [4096,256] @ [256,32000] -> d_out
  gemm_kernel<<<dim3(VV / 128, MM / 128), 256, 0, stream>>>(
      x, logitW, nullptr, nullptr, out, MM, VV, DD, 0);
}